// EMLPBlock_6416681140283
// MI455X (gfx1250) — compile-verified
//
#include <hip/hip_runtime.h>
#include <hip/hip_bf16.h>

// Problem sizes (from reference)
#define BB    4096
#define NIN   512
#define NREP  512
#define NOUT  640
#define RW    512
#define RB    64
#define NNZ   16384

typedef __attribute__((ext_vector_type(2))) float v2f;
typedef __attribute__((ext_vector_type(8))) float v8f;

// Workspace layout (floats): c[512] | bp[640] | Wp[NOUT*NIN]
#define WS_C   0
#define WS_BP  512
#define WS_WP  1152

// ---------------------------------------------------------------------------
// Kernel 0: zero the atomic accumulator c (must be re-zeroed every call)
// ---------------------------------------------------------------------------
__global__ void k_zero_c(float* __restrict__ ws) {
    int t = threadIdx.x;
    ws[WS_C + t]       = 0.0f;
    ws[WS_C + t + 256] = 0.0f;
}

// ---------------------------------------------------------------------------
// Kernel 1: c = Qw^T @ w_flat   (Qw stream #1: b128 loads, LDS partials,
//                                one global atomic per column per block)
// grid = 640 blocks x 256 threads; each block owns 512 rows of Qw.
// Thread t handles columns [4*(t&127), +4) of rows r ≡ (t>>7) (mod 2).
// ---------------------------------------------------------------------------
__global__ void k_qwt_w(const float* __restrict__ Qw,
                        const float* __restrict__ wflat,
                        float* __restrict__ ws) {
    __shared__ float part[RW];
    const int t = threadIdx.x;
    part[t]       = 0.0f;
    part[t + 256] = 0.0f;
    __syncthreads();

    const int half = t >> 7;          // 0 or 1 (wave-uniform: 32 | 128)
    const int c4   = (t & 127) * 4;   // column base (16B aligned)
    const size_t r0 = (size_t)blockIdx.x * 512;

    float4 acc = {0.f, 0.f, 0.f, 0.f};
    for (int r = half; r < 512; r += 2) {
        const float wv = wflat[r0 + r];                     // scalar load
        const float4 q = *(const float4*)(Qw + (r0 + r) * RW + c4);
        acc.x += q.x * wv;
        acc.y += q.y * wv;
        acc.z += q.z * wv;
        acc.w += q.w * wv;
    }
    atomicAdd(&part[c4 + 0], acc.x);   // ds_add_f32, 2-way sharing only
    atomicAdd(&part[c4 + 1], acc.y);
    atomicAdd(&part[c4 + 2], acc.z);
    atomicAdd(&part[c4 + 3], acc.w);
    __syncthreads();
    atomicAdd(&ws[WS_C + t],       part[t]);
    atomicAdd(&ws[WS_C + t + 256], part[t + 256]);
}

// ---------------------------------------------------------------------------
// Kernel 2: bp = Qb @ (Qb^T @ b)   (tiny; single block)
// ---------------------------------------------------------------------------
__global__ void k_qb_proj(const float* __restrict__ Qb,
                          const float* __restrict__ b,
                          float* __restrict__ ws) {
    __shared__ float bs[NOUT];
    __shared__ float cb[RB];
    const int t = threadIdx.x;
    for (int o = t; o < NOUT; o += 256) bs[o] = b[o];
    __syncthreads();
    if (t < RB) {
        float a = 0.0f;
        for (int o = 0; o < NOUT; ++o) a += Qb[(size_t)o * RB + t] * bs[o];
        cb[t] = a;
    }
    __syncthreads();
    for (int o = t; o < NOUT; o += 256) {
        float a = 0.0f;
        const float* q = Qb + (size_t)o * RB;
        #pragma unroll
        for (int r = 0; r < RB; ++r) a += q[r] * cb[r];
        ws[WS_BP + o] = a;
    }
}

// ---------------------------------------------------------------------------
// Kernel 3: Wp[i] = dot(Qw[i, :], c)   (Qw stream #2: b128 loads)
// 8 waves/block, each wave does 16 rows: 4x global_load_b128 + 4x ds b128
// per row per lane, then shuffle-reduce. grid = 2560 blocks (128 rows each).
// ---------------------------------------------------------------------------
__global__ void k_make_wp(const float* __restrict__ Qw,
                          float* __restrict__ ws) {
    __shared__ float cs[RW];
    const int t = threadIdx.x;
    cs[t]       = ws[WS_C + t];
    cs[t + 256] = ws[WS_C + t + 256];
    __syncthreads();
    const int w    = t >> 5;
    const int lane = t & 31;
    for (int j = 0; j < 16; ++j) {
        const size_t i = (size_t)blockIdx.x * 128 + (size_t)j * 8 + w;
        const float* q = Qw + i * RW;
        float acc = 0.0f;
        #pragma unroll
        for (int s = 0; s < 4; ++s) {
            const int idx = lane * 4 + 128 * s;     // 16B aligned, no conflicts
            const float4 qv = *(const float4*)(q  + idx);
            const float4 cv = *(const float4*)(cs + idx);
            acc += qv.x * cv.x + qv.y * cv.y + qv.z * cv.z + qv.w * cv.w;
        }
        #pragma unroll
        for (int off = 16; off > 0; off >>= 1)
            acc += __shfl_xor(acc, off, 32);
        if (lane == 0) ws[WS_WP + i] = acc;
    }
}

// ---------------------------------------------------------------------------
// Kernel 4 (fused): per 32-row batch tile
//   lin = x @ Wp^T + bp   via v_wmma_f32_16x16x4_f32  (f32 WMMA)
//   preact = lin + 0.1 * bilinear_scatter(lin)        (LDS atomics)
//   out = sigmoid(preact[:, gate_idx]) * preact[:, :NREP]
// LDS: xs[32][516] + lin[32][641] + pre[32][641]  (padded strides: 641 odd, 516 ≡ 4 mod 64 banks)
// ---------------------------------------------------------------------------
#define XS_STRIDE 516
#define LN_STRIDE 641
#define SMEM_FLOATS (32 * XS_STRIDE + 2 * 32 * LN_STRIDE)

__global__ void k_fused(const float* __restrict__ x,
                        const float* __restrict__ ws,       // bp, Wp
                        const float* __restrict__ bi_params,
                        const int*   __restrict__ bi_src,
                        const int*   __restrict__ bi_row,
                        const int*   __restrict__ bi_col,
                        const int*   __restrict__ gate_idx,
                        float* __restrict__ out) {
    extern __shared__ float smem[];
    float* xs  = smem;                       // 32 x 516
    float* lin = smem + 32 * XS_STRIDE;      // 32 x 641
    float* pre = lin  + 32 * LN_STRIDE;      // 32 x 641

    const float* Wp = ws + WS_WP;
    const float* bp = ws + WS_BP;

    const int t    = threadIdx.x;
    const int w    = t >> 5;
    const int lane = t & 31;
    const int row0 = blockIdx.x * 32;

    // ---- stage x tile into LDS (coalesced) ----
    for (int idx = t; idx < 32 * NIN; idx += 256) {
        const int r = idx >> 9;        // /512
        const int c = idx & 511;
        xs[r * XS_STRIDE + c] = x[(size_t)(row0 + r) * NIN + c];
    }
    __syncthreads();

    // ---- WMMA GEMM: 2 (M) x 40 (N) subtiles of 16x16, 8 waves round-robin ----
    const int halfK = (lane >> 4) << 1;    // 0 or 2
    const int nlane = lane & 15;
    for (int tile = w; tile < 80; tile += 8) {
        const int m0 = (tile & 1) * 16;
        const int n0 = (tile >> 1) * 16;
        v8f acc = {0.f, 0.f, 0.f, 0.f, 0.f, 0.f, 0.f, 0.f};
        const float* xrow = xs + (m0 + nlane) * XS_STRIDE;
        const float* wrow = Wp + (size_t)(n0 + nlane) * NIN;
        #pragma unroll 4
        for (int k0 = 0; k0 < NIN; k0 += 4) {
            v2f a  = *(const v2f*)(xrow + k0 + halfK);
            v2f bf = *(const v2f*)(wrow + k0 + halfK);
            acc = __builtin_amdgcn_wmma_f32_16x16x4_f32(
                    false, a, false, bf, (short)0, acc, false, false);
        }
        const int nIdx  = n0 + nlane;
        const float bpn = bp[nIdx];
        const int mBase = m0 + ((lane >> 4) << 3);   // +8 for upper half-lanes
        #pragma unroll
        for (int v = 0; v < 8; ++v)
            lin[(mBase + v) * LN_STRIDE + nIdx] = acc[v] + bpn;
    }
    __syncthreads();

    // ---- preact = lin (copy) ----
    for (int r = 0; r < 32; ++r)
        for (int c = t; c < NOUT; c += 256)
            pre[r * LN_STRIDE + c] = lin[r * LN_STRIDE + c];
    __syncthreads();

    // ---- bilinear: lane owns batch row `lane`; wave w takes nnz k = w + 8j
    //      (k wave-uniform -> scalar index loads; rows distinct per lane) ----
    {
        const float* lrow = lin + lane * LN_STRIDE;
        float* prow       = pre + lane * LN_STRIDE;
        for (int k = w; k < NNZ; k += 8) {
            const int ku = __builtin_amdgcn_readfirstlane(k);
            const int s  = __builtin_amdgcn_readfirstlane(bi_src[ku]);
            const int rI = __builtin_amdgcn_readfirstlane(bi_row[ku]);
            const int cI = __builtin_amdgcn_readfirstlane(bi_col[ku]);
            const float p = bi_params[ku];
            const float val = 0.1f * p * lrow[s] * lrow[cI];
            atomicAdd(&prow[rI], val);   // ds_add_f32
        }
    }
    __syncthreads();

    // ---- gated sigmoid + coalesced store ----
    for (int idx = t; idx < 32 * NREP; idx += 256) {
        const int r = idx >> 9;        // /512
        const int j = idx & 511;
        const float pa = pre[r * LN_STRIDE + j];
        const int   g  = gate_idx[j];
        const float pg = pre[r * LN_STRIDE + g];
        const float sg = 1.0f / (1.0f + __expf(-pg));
        out[(size_t)(row0 + r) * NREP + j] = sg * pa;
    }
}

// ---------------------------------------------------------------------------
extern "C" void kernel_launch(void* const* d_in, const int* in_sizes, int n_in,
                              void* d_out, int out_size, void* d_ws, size_t ws_size,
                              hipStream_t stream) {
    const float* x         = (const float*)d_in[0];
    const float* W_weight  = (const float*)d_in[1];
    const float* b         = (const float*)d_in[2];
    const float* Qw        = (const float*)d_in[3];
    const float* Qb        = (const float*)d_in[4];
    const float* bi_params = (const float*)d_in[5];
    const int*   bi_src    = (const int*)d_in[6];
    const int*   bi_row    = (const int*)d_in[7];
    const int*   bi_col    = (const int*)d_in[8];
    const int*   gate_idx  = (const int*)d_in[9];
    float* out = (float*)d_out;
    float* ws  = (float*)d_ws;

    // c = 0
    k_zero_c<<<1, 256, 0, stream>>>(ws);
    // c = Qw^T @ w  (671 MB stream #1, b128 loads)
    k_qwt_w<<<(NOUT * NIN) / 512, 256, 0, stream>>>(Qw, W_weight, ws);
    // bp = Qb Qb^T b
    k_qb_proj<<<1, 256, 0, stream>>>(Qb, b, ws);
    // Wp = Qw @ c   (671 MB stream #2, b128 loads)
    k_make_wp<<<(NOUT * NIN) / 128, 256, 0, stream>>>(Qw, ws);
    // fused WMMA GEMM + bilinear + gate, 32 batch rows per block
    k_fused<<<BB / 32, 256, SMEM_FLOATS * sizeof(float), stream>>>(
        x, ws, bi_params, bi_src, bi_row, bi_col, gate_idx, out);
}